// OneHotEmbeddings_51608327029374
// MI455X (gfx1250) — compile-verified
//
#include <hip/hip_runtime.h>

// One-hot embedding == gather: out[t, h] = W[h * VOCAB + ids[t]] + b[h]
// ids: int32 [4*2048], W: f32 [1024, 32000] row-major, b: f32 [1024]
// out: f32 [4*2048, 1024]

typedef float v4f __attribute__((ext_vector_type(4)));

constexpr int VOCAB       = 32000;
constexpr int HIDDEN      = 1024;
constexpr int NTOK        = 4 * 2048;
constexpr int TPB         = 256;   // 8 waves (wave32) per block
constexpr int TOK_PER_BLK = 4;

__global__ __launch_bounds__(TPB) void onehot_embed_gather_kernel(
    const int*   __restrict__ ids,
    const float* __restrict__ W,
    const float* __restrict__ bias,
    float*       __restrict__ out)
{
    __shared__ float s_bias[HIDDEN];

    const int tid = threadIdx.x;
    const int h0  = tid * 4;            // this thread's 4 hidden dims

    // ---- Stage the 4KB bias vector into LDS via the CDNA5 async copy path.
    // Each thread async-copies its own 16B slice and later reads back the
    // same 16B, so a wave-local s_wait_asynccnt is sufficient for the RAW.
    {
        const float* gsrc     = bias + h0;
        unsigned     lds_addr = (unsigned)(unsigned long long)(&s_bias[h0]);
        asm volatile("global_load_async_to_lds_b128 %0, %1, off"
                     :: "v"(lds_addr), "v"(gsrc) : "memory");
        asm volatile("s_wait_asynccnt 0" ::: "memory");
    }
    __syncthreads();

    const int t0 = blockIdx.x * TOK_PER_BLK;

    // Block-uniform token ids -> scalar loads (KMcnt path).
    int id[TOK_PER_BLK];
#pragma unroll
    for (int j = 0; j < TOK_PER_BLK; ++j) {
        id[j] = ids[t0 + j];
    }

    const v4f b4 = *(const v4f*)(&s_bias[h0]);

    // 16 independent strided gather loads per lane (4 tokens x 4 rows),
    // issued back-to-back for memory-level parallelism. Rows are 128KB
    // apart so these cannot be vectorized; lines are reused ~8x across
    // tokens landing in the same 32-id vocab bucket (L2-resident W).
    v4f acc[TOK_PER_BLK];
#pragma unroll
    for (int j = 0; j < TOK_PER_BLK; ++j) {
        const float* col = W + id[j];
        acc[j].x = col[(h0 + 0) * VOCAB];
        acc[j].y = col[(h0 + 1) * VOCAB];
        acc[j].z = col[(h0 + 2) * VOCAB];
        acc[j].w = col[(h0 + 3) * VOCAB];
    }

    // Coalesced b128 stores; non-temporal so the streamed 33.5MB output
    // does not evict the W working set (131MB) from the 192MB L2.
#pragma unroll
    for (int j = 0; j < TOK_PER_BLK; ++j) {
        v4f r = acc[j] + b4;
        __builtin_nontemporal_store(
            r, (v4f*)(out + (size_t)(t0 + j) * HIDDEN + h0));
    }
}

extern "C" void kernel_launch(void* const* d_in, const int* in_sizes, int n_in,
                              void* d_out, int out_size, void* d_ws, size_t ws_size,
                              hipStream_t stream)
{
    const int*   ids  = (const int*)  d_in[0];   // input_ids, int32, 8192
    const float* W    = (const float*)d_in[1];   // [1024, 32000]
    const float* bias = (const float*)d_in[2];   // [1024]
    float*       out  = (float*)      d_out;     // [8192, 1024]

    const int nblocks = NTOK / TOK_PER_BLK;      // 2048
    onehot_embed_gather_kernel<<<nblocks, TPB, 0, stream>>>(ids, W, bias, out);
}